// RoI_71923522339118
// MI455X (gfx1250) — compile-verified
//
#include <hip/hip_runtime.h>
#include <stdint.h>

#define B_ 4
#define C_ 256
#define H_ 128
#define W_ 128
#define Q_ 100
#define OUT_ 7
#define STRIDE_ 129   // 128 + 1 pad word -> bank-conflict-free column pass

#if defined(__gfx1250__) && __has_builtin(__builtin_amdgcn_tensor_load_to_lds) && __has_builtin(__builtin_amdgcn_s_wait_tensorcnt)
#define USE_TDM 1
#else
#define USE_TDM 0
#endif

typedef unsigned int u32;
typedef u32 v4u __attribute__((ext_vector_type(4)));
typedef int  v8i __attribute__((ext_vector_type(8)));
typedef int  v4i __attribute__((ext_vector_type(4)));

__device__ __forceinline__ float wave_incl_scan(float v, int lane) {
#pragma unroll
  for (int d = 1; d < 32; d <<= 1) {
    float t = __shfl_up(v, (unsigned)d, 32);
    if (lane >= d) v += t;
  }
  return v;
}

__global__ __launch_bounds__(256) void roi_sat_pool_kernel(
    const float* __restrict__ x, const int* __restrict__ rois,
    float* __restrict__ out)
{
  __shared__ float tile[H_ * STRIDE_];   // 66,048 B, padded SAT plane
  __shared__ int   roiLds[Q_ * 5];       // 2,000 B

  const int tid  = threadIdx.x;
  const int lane = tid & 31;
  const int wave = tid >> 5;
  const int bc   = blockIdx.x;           // bc = b*C + c
  const int b    = bc / C_;

  const float* plane = x + (size_t)bc * (H_ * W_);

  // ---------------- Phase 1: global -> LDS (TDM, one DMA per workgroup) ---
#if USE_TDM
  if (wave == 0) {
    u32 lds_base = (u32)(uintptr_t)(&tile[0]);            // flat addr low 32 = LDS offset
    unsigned long long ga = (unsigned long long)(uintptr_t)plane;
    // D# group 0: count=1, lds_addr, global_addr[56:0], type=2 (bits 127:126)
    v4u g0 = { 1u,
               lds_base,
               (u32)(ga & 0xFFFFFFFFull),
               (u32)(((ga >> 32) & 0x01FFFFFFull) | 0x80000000u) };
    // D# group 1:
    // dw0: data_size=2(4B)<<16 | pad_enable<<20 | pad_interval=6(128DW)<<22, pad_amount=0(1DW)
    // dw1: tensor_dim0(=128)<<16 ; dw2: tensor_dim1(=128)<<16 ; dw3: tile_dim0(=128)<<16
    // dw4: tile_dim1=128 (tile_dim2=0) ; dw5: tensor_dim0_stride=128 ; dw6,dw7: 0
    v8i g1 = { (int)0x01920000u,
               (int)(128u << 16),
               (int)(128u << 16),
               (int)(128u << 16),
               128,
               128,
               0, 0 };
    v4i g2 = {0, 0, 0, 0};   // dims 2..4 unused (tile_dim2 == 0)
    v4i g3 = {0, 0, 0, 0};
    v8i g4 = {0, 0, 0, 0, 0, 0, 0, 0};
    __builtin_amdgcn_tensor_load_to_lds(g0, g1, g2, g3, g4, 0);
  }
#else
  for (int idx = tid; idx < H_ * W_; idx += 256)
    tile[(idx >> 7) * STRIDE_ + (idx & 127)] = plane[idx];
#endif

  // overlap: stage this batch's ROIs into LDS while the TDM runs
  for (int i = tid; i < Q_ * 5; i += 256)
    roiLds[i] = rois[b * (Q_ * 5) + i];

#if USE_TDM
  if (wave == 0) __builtin_amdgcn_s_wait_tensorcnt(0);
#endif
  __syncthreads();

  // ---------------- Phase 2a: inclusive scan along rows (W axis) ----------
  for (int r = wave; r < H_; r += 8) {
    float* row = &tile[r * STRIDE_ + lane * 4];
    float a0 = row[0], a1 = row[1], a2 = row[2], a3 = row[3];
    a1 += a0; a2 += a1; a3 += a2;
    float excl = wave_incl_scan(a3, lane) - a3;
    row[0] = a0 + excl; row[1] = a1 + excl; row[2] = a2 + excl; row[3] = a3 + excl;
  }
  __syncthreads();

  // ---------------- Phase 2b: inclusive scan along columns (H axis) -------
  for (int col = wave; col < W_; col += 8) {
    float* p = &tile[(lane * 4) * STRIDE_ + col];
    float a0 = p[0], a1 = p[STRIDE_], a2 = p[2 * STRIDE_], a3 = p[3 * STRIDE_];
    a1 += a0; a2 += a1; a3 += a2;
    float excl = wave_incl_scan(a3, lane) - a3;
    p[0] = a0 + excl; p[STRIDE_] = a1 + excl;
    p[2 * STRIDE_] = a2 + excl; p[3 * STRIDE_] = a3 + excl;
  }
  __syncthreads();

  // ---------------- Phase 3: evaluate all (q, i, j) bins from the SAT -----
  const int total = Q_ * OUT_ * OUT_;      // 4900
  for (int t = tid; t < total; t += 256) {
    int q  = t / 49;
    int ij = t - q * 49;
    int i  = ij / 7;
    int j  = ij - i * 7;
    const int* r = &roiLds[q * 5];         // [bidx, x1, y1, x2, y2]
    int x1 = r[1], y1 = r[2], x2 = r[3], y2 = r[4];
    int lw = x2 - x1, lh = y2 - y1;
    // reference: lo = s + floor(k*L/7); hi = s + ceil((k+1)*L/7)  (all args >= 0)
    int xs = x1 + (j * lw) / 7;
    int xe = x1 + ((j + 1) * lw + 6) / 7;
    int ys = y1 + (i * lh) / 7;
    int ye = y1 + ((i + 1) * lh + 6) / 7;

    float sSE = tile[(ye - 1) * STRIDE_ + (xe - 1)];
    float sNE = (ys > 0)            ? tile[(ys - 1) * STRIDE_ + (xe - 1)] : 0.0f;
    float sSW = (xs > 0)            ? tile[(ye - 1) * STRIDE_ + (xs - 1)] : 0.0f;
    float sNW = (ys > 0 && xs > 0)  ? tile[(ys - 1) * STRIDE_ + (xs - 1)] : 0.0f;
    float s    = sSE - sNE - sSW + sNW;
    float area = (float)((ye - ys) * (xe - xs));
    out[(size_t)((b * Q_ + q) * C_ + bc % C_) * 49 + ij] = s / area;
  }
}

extern "C" void kernel_launch(void* const* d_in, const int* in_sizes, int n_in,
                              void* d_out, int out_size, void* d_ws, size_t ws_size,
                              hipStream_t stream) {
  const float* x    = (const float*)d_in[0];
  const int*   rois = (const int*)d_in[1];
  float*       out  = (float*)d_out;
  dim3 grid(B_ * C_);
  dim3 block(256);
  hipLaunchKernelGGL(roi_sat_pool_kernel, grid, block, 0, stream, x, rois, out);
}